// CrossAttention_DenseAVInteractions_27444841022212
// MI455X (gfx1250) — compile-verified
//
#include <hip/hip_runtime.h>
#include <hip/hip_bf16.h>

// ============================================================================
// CrossAttention with dense AV interactions, factorized.
//
// Key identity: k[(i,j)] = pv_k[i] + pa_k[j], v[(i,j)] = vv[i] + va[j]
//   => softmax over the 65536 (i,j) grid factorizes into two independent
//      256-way softmaxes, and  out = softmax(Sv)@Vv + softmax(Sa)@Va.
// Exact in real arithmetic; computed in f32 with V_WMMA_F32_16X16X4_F32.
// ============================================================================

typedef __attribute__((ext_vector_type(2))) float v2f;
typedef __attribute__((ext_vector_type(8))) float v8f;

#define DIM 512
#define HD  64
#define NH  8

// Padded LDS row strides: stride % 64 == 4 makes per-lane bank index
// (4*lr + ka) mod 64 -> conflict-free ds_load_b64 fragment fetches.
#define KQ_STRIDE 68    // for q_s and kbuf (row length 64)
#define AL_STRIDE 260   // for alpha and vT (row length 256)

// ---------------------------------------------------------------------------
// f32 WMMA 16x16x4 wrapper.
// Fragment layouts (CDNA5 ISA 7.12.2, 32-bit data):
//   A (16x4, MxK): lane l, vgpr r -> A[M = l%16][K = 2*(l/16) + r]
//   B (4x16, KxN): lane l, vgpr r -> B[K = 2*(l/16) + r][N = l%16]
//   C/D (16x16):   lane l, vgpr r -> C[M = r + 8*(l/16)][N = l%16]
// ---------------------------------------------------------------------------
__device__ inline v8f wmma4(v2f a, v2f b, v8f c) {
  return __builtin_amdgcn_wmma_f32_16x16x4_f32(
      /*neg_a=*/false, a, /*neg_b=*/false, b,
      /*c_mod=*/(short)0, c, /*reuse_a=*/false, /*reuse_b=*/false);
}

__device__ inline v2f ld2(const float* p) { return *(const v2f*)p; }  // 8B-aligned

// ---------------------------------------------------------------------------
// Generic C[m][n] = alphaS * sum_k A[m][k] * Bw[n][bkoff + k]  (+ bias[n])
// i.e. X @ W^T with torch-style [out,in] weights. One wave computes a 16x64
// output strip (4 N-tiles, A-fragment reused 4x). M%16==0, N%64==0, K%4==0.
// 1-deep software pipeline: next k-step's fragments are fetched before the
// current k-step's WMMAs, so waits relax from loadcnt==0 to loadcnt<=5.
// ---------------------------------------------------------------------------
__global__ __launch_bounds__(128)
void gemm_bt_wmma(const float* __restrict__ A, int lda,
                  const float* __restrict__ Bw, int ldb, int bkoff,
                  float* __restrict__ C, int ldc,
                  int M, int N, int K, float alphaS,
                  const float* __restrict__ bias)
{
  const int wid = (blockIdx.x * blockDim.x + threadIdx.x) >> 5;
  const int ngrp = N >> 6;                 // groups of 4 tiles along N
  const int total = (M >> 4) * ngrp;
  if (wid >= total) return;                // wave-uniform
  const int tm = wid / ngrp;
  const int tg = wid % ngrp;
  const int lane = threadIdx.x & 31;
  const int lr = lane & 15;
  const int hi = lane >> 4;

  const float* arow  = A  + (size_t)(tm * 16 + lr) * lda + hi * 2;
  const float* brow0 = Bw + (size_t)(tg * 64 +  0 + lr) * ldb + bkoff + hi * 2;
  const float* brow1 = Bw + (size_t)(tg * 64 + 16 + lr) * ldb + bkoff + hi * 2;
  const float* brow2 = Bw + (size_t)(tg * 64 + 32 + lr) * ldb + bkoff + hi * 2;
  const float* brow3 = Bw + (size_t)(tg * 64 + 48 + lr) * ldb + bkoff + hi * 2;

  v8f acc0 = {0,0,0,0,0,0,0,0}, acc1 = {0,0,0,0,0,0,0,0};
  v8f acc2 = {0,0,0,0,0,0,0,0}, acc3 = {0,0,0,0,0,0,0,0};

  // prologue
  v2f a  = ld2(arow);
  v2f b0 = ld2(brow0);
  v2f b1 = ld2(brow1);
  v2f b2 = ld2(brow2);
  v2f b3 = ld2(brow3);

#pragma unroll 4
  for (int k0 = 4; k0 < K; k0 += 4) {
    // prefetch next k-step while current WMMAs run
    const v2f an  = ld2(arow  + k0);
    const v2f bn0 = ld2(brow0 + k0);
    const v2f bn1 = ld2(brow1 + k0);
    const v2f bn2 = ld2(brow2 + k0);
    const v2f bn3 = ld2(brow3 + k0);
    acc0 = wmma4(a, b0, acc0);
    acc1 = wmma4(a, b1, acc1);
    acc2 = wmma4(a, b2, acc2);
    acc3 = wmma4(a, b3, acc3);
    a = an; b0 = bn0; b1 = bn1; b2 = bn2; b3 = bn3;
  }
  // epilogue
  acc0 = wmma4(a, b0, acc0);
  acc1 = wmma4(a, b1, acc1);
  acc2 = wmma4(a, b2, acc2);
  acc3 = wmma4(a, b3, acc3);

  v8f* accs[4] = {&acc0, &acc1, &acc2, &acc3};
#pragma unroll
  for (int t = 0; t < 4; ++t) {
    const int n = tg * 64 + t * 16 + lr;
    const float bv = bias ? bias[n] : 0.0f;
#pragma unroll
    for (int r = 0; r < 8; ++r) {
      const int m = tm * 16 + r + 8 * hi;
      C[(size_t)m * ldc + n] = alphaS * (*accs[t])[r] + bv;
    }
  }
}

// ---------------------------------------------------------------------------
// Fused factorized attention. One workgroup (4 waves) per (b,h).
// Wave w owns query rows [16w, 16w+16).
//   Qbuf: [B,64,512] (pre-scaled by hd^-0.5), head h slice = cols h*64..
//   Pv/Pa: [B,256,1024]; k part = cols h*64.., v part = cols 512+h*64..
//   Obuf: [B,64,512] attention output (pre-projection), head-interleaved.
// LDS: q_s[64][68], kbuf[256][68] (reused as alpha[64][260]), vT[64][260].
// ---------------------------------------------------------------------------
__global__ __launch_bounds__(128)
void attn_fused(const float* __restrict__ Qbuf,
                const float* __restrict__ Pv,
                const float* __restrict__ Pa,
                float* __restrict__ Obuf)
{
  __shared__ float q_s[64 * KQ_STRIDE];    // 17.0 KB : scaled Q head tile
  __shared__ float kbuf[256 * KQ_STRIDE];  // 68.0 KB : K rows, reused as alpha
  __shared__ float vT[64 * AL_STRIDE];     // 65.0 KB : V transposed [d][key]

  const int b = blockIdx.x >> 3;
  const int h = blockIdx.x & 7;
  const int tid = threadIdx.x;
  const int w = tid >> 5;
  const int lane = tid & 31;
  const int lr = lane & 15;
  const int hi = lane >> 4;

  // stage Q (already scaled)
  const float* Qb = Qbuf + (size_t)(b * 64) * DIM + h * HD;
  for (int i = tid; i < 64 * 64; i += 128)
    q_s[(i >> 6) * KQ_STRIDE + (i & 63)] = Qb[(i >> 6) * DIM + (i & 63)];

  v8f o0 = {0,0,0,0,0,0,0,0}, o1 = {0,0,0,0,0,0,0,0};
  v8f o2 = {0,0,0,0,0,0,0,0}, o3 = {0,0,0,0,0,0,0,0};
  v8f* outs[4] = {&o0, &o1, &o2, &o3};

  for (int branch = 0; branch < 2; ++branch) {
    const float* Pb = (branch ? Pa : Pv) + (size_t)(b * 256) * 1024;

    __syncthreads();  // prior branch done reading kbuf/vT
    for (int i = tid; i < 256 * 64; i += 128) {
      const int r = i >> 6, c = i & 63;
      kbuf[r * KQ_STRIDE + c] = Pb[r * 1024 + h * HD + c];        // K rows
      vT[c * AL_STRIDE + r]   = Pb[r * 1024 + 512 + h * HD + c];  // V^T
    }
    __syncthreads();  // also guards q_s on first branch

    // ---- S = Qh @ Kh^T : 16 column tiles of 16x16, K=64 in steps of 4 ----
    // Fetch all 17 fragments of a k-step before its 16 WMMAs so the
    // ds_loads batch under one relaxed s_wait_dscnt.
    v8f acc[16];
#pragma unroll
    for (int nt = 0; nt < 16; ++nt) acc[nt] = (v8f){0,0,0,0,0,0,0,0};

    const float* qrow = q_s + (16 * w + lr) * KQ_STRIDE + hi * 2;
    for (int k0 = 0; k0 < 64; k0 += 4) {
      const v2f a = ld2(qrow + k0);
      v2f bb[16];
#pragma unroll
      for (int nt = 0; nt < 16; ++nt)
        bb[nt] = ld2(kbuf + (16 * nt + lr) * KQ_STRIDE + hi * 2 + k0);
#pragma unroll
      for (int nt = 0; nt < 16; ++nt)
        acc[nt] = wmma4(a, bb[nt], acc[nt]);
    }

    // ---- row softmax (row m = 16w + r + 8*hi; 16 lanes share a row) ----
    float rcp[8];
#pragma unroll
    for (int r = 0; r < 8; ++r) {
      float mx = -3.402823466e38f;
#pragma unroll
      for (int nt = 0; nt < 16; ++nt) mx = fmaxf(mx, acc[nt][r]);
      for (int d = 1; d < 16; d <<= 1) mx = fmaxf(mx, __shfl_xor(mx, d, 32));
      float s = 0.0f;
#pragma unroll
      for (int nt = 0; nt < 16; ++nt) {
        const float e = __expf(acc[nt][r] - mx);
        acc[nt][r] = e;
        s += e;
      }
      for (int d = 1; d < 16; d <<= 1) s += __shfl_xor(s, d, 32);
      rcp[r] = 1.0f / s;
    }

    __syncthreads();  // every wave done reading kbuf for S

    // ---- C-layout -> A-layout via LDS: alpha[64][260] (unnormalized) ----
    float* alpha = kbuf;  // 64*260 floats fit in kbuf (256*68)
#pragma unroll
    for (int nt = 0; nt < 16; ++nt)
#pragma unroll
      for (int r = 0; r < 8; ++r)
        alpha[(16 * w + r + 8 * hi) * AL_STRIDE + 16 * nt + lr] = acc[nt][r];
    // wave reads only its own rows; LDS ops are in-order within a wave.

    // ---- O_branch = alpha @ V ; fold 1/rowsum into the accumulation ----
    const float* arow = alpha + (16 * w + lr) * AL_STRIDE + hi * 2;
#pragma unroll
    for (int t = 0; t < 4; ++t) {
      const float* vrow = vT + (16 * t + lr) * AL_STRIDE + hi * 2;
      v8f oc = {0,0,0,0,0,0,0,0};
      // 1-deep pipeline over k-steps
      v2f a  = ld2(arow);
      v2f bb = ld2(vrow);
#pragma unroll 4
      for (int k0 = 4; k0 < 256; k0 += 4) {
        const v2f an  = ld2(arow + k0);
        const v2f bbn = ld2(vrow + k0);
        oc = wmma4(a, bb, oc);
        a = an; bb = bbn;
      }
      oc = wmma4(a, bb, oc);
#pragma unroll
      for (int r = 0; r < 8; ++r) (*outs[t])[r] += oc[r] * rcp[r];
    }
  }

  // ---- write head output: Obuf[b*64 + row][h*64 + col] ----
#pragma unroll
  for (int t = 0; t < 4; ++t)
#pragma unroll
    for (int r = 0; r < 8; ++r) {
      const int row = 16 * w + r + 8 * hi;
      const int col = 16 * t + lr;
      Obuf[(size_t)(b * 64 + row) * DIM + h * HD + col] = (*outs[t])[r];
    }
}

// ---------------------------------------------------------------------------
// Launch
// ---------------------------------------------------------------------------
extern "C" void kernel_launch(void* const* d_in, const int* in_sizes, int n_in,
                              void* d_out, int out_size, void* d_ws, size_t ws_size,
                              hipStream_t stream) {
  (void)n_in; (void)out_size; (void)ws_size;
  const float* xmm   = (const float*)d_in[0];  // [B,64,512]
  const float* xa    = (const float*)d_in[1];  // [B,256,512]
  const float* xv    = (const float*)d_in[2];  // [B,256,512]
  const float* Wq    = (const float*)d_in[3];  // [512,512]
  const float* Wkv   = (const float*)d_in[4];  // [1024,1024]
  const float* Wproj = (const float*)d_in[5];  // [512,512]
  const float* bproj = (const float*)d_in[6];  // [512]
  float* out = (float*)d_out;                  // [B,64,512]

  const int B = in_sizes[0] / (64 * DIM);
  float* ws   = (float*)d_ws;
  float* Qbuf = ws;                               // B*64*512
  float* Pv   = Qbuf + (size_t)B * 64 * DIM;      // B*256*1024
  float* Pa   = Pv   + (size_t)B * 256 * 1024;    // B*256*1024
  float* Obuf = Pa   + (size_t)B * 256 * 1024;    // B*64*512

  const float scale = 0.125f;  // hd^-0.5, hd = 64

  // 1) projections: Q (pre-scaled), Pv = xv@Wkv[:, :512]^T, Pa = xa@Wkv[:, 512:]^T
  {
    int waves = ((B * 64) / 16) * (DIM / 64);
    gemm_bt_wmma<<<waves / 4, 128, 0, stream>>>(xmm, DIM, Wq, DIM, 0,
                                                Qbuf, DIM, B * 64, DIM, DIM,
                                                scale, nullptr);
  }
  {
    int waves = ((B * 256) / 16) * (1024 / 64);
    gemm_bt_wmma<<<waves / 4, 128, 0, stream>>>(xv, DIM, Wkv, 1024, 0,
                                                Pv, 1024, B * 256, 1024, DIM,
                                                1.0f, nullptr);
    gemm_bt_wmma<<<waves / 4, 128, 0, stream>>>(xa, DIM, Wkv, 1024, DIM,
                                                Pa, 1024, B * 256, 1024, DIM,
                                                1.0f, nullptr);
  }

  // 2) fused factorized attention: one block per (b,h)
  attn_fused<<<B * NH, 128, 0, stream>>>(Qbuf, Pv, Pa, Obuf);

  // 3) output projection with bias
  {
    int waves = ((B * 64) / 16) * (DIM / 64);
    gemm_bt_wmma<<<waves / 4, 128, 0, stream>>>(Obuf, DIM, Wproj, DIM, 0,
                                                out, DIM, B * 64, DIM, DIM,
                                                1.0f, bproj);
  }
}